// TransformerLayer_32899449487822
// MI455X (gfx1250) — compile-verified
//
#include <hip/hip_runtime.h>
#include <hip/hip_bf16.h>

typedef __attribute__((ext_vector_type(2))) float v2f;
typedef __attribute__((ext_vector_type(8))) float v8f;

#define SEQ_LEN 2048
#define BATCH   32
#define MLP_HID 256

__global__ __launch_bounds__(1024)
void fused_attn_mlp_gfx1250(const float* __restrict__ src,
                            const float* __restrict__ Wq, const float* __restrict__ bq,
                            const float* __restrict__ Wk, const float* __restrict__ bk,
                            const float* __restrict__ Wv, const float* __restrict__ bv,
                            const float* __restrict__ bias,
                            const float* __restrict__ W1, const float* __restrict__ b1,
                            const float* __restrict__ W2, const float* __restrict__ b2,
                            float* __restrict__ out)
{
    __shared__ float s_ctx[BATCH];

    const int tid  = (int)threadIdx.x;
    const int lane = tid & 31;
    const int wave = tid >> 5;          // 32 waves; wave w owns batch w

    // ---------------- Phase 1: attention row s=0, online softmax ----------------
    {
        const int   b   = wave;
        const float wk  = Wk[0], wv = Wv[0];
        const float bkv = bk[0], bvv = bv[0];
        const float qb  = Wq[0] * src[b] + bq[0];      // src[0*BATCH + b]

        float m = -INFINITY, l = 0.f, acc = 0.f;
        #pragma unroll 4
        for (int t = lane; t < SEQ_LEN; t += 32) {
            const float x  = src[t * BATCH + b];
            const float s  = qb * (wk * x + bkv) + bias[t];   // bias[0, 0, t]
            const float v  = wv * x + bvv;
            const float nm = fmaxf(m, s);
            const float sc = __expf(m - nm);
            const float p  = __expf(s - nm);
            l   = l   * sc + p;
            acc = acc * sc + p * v;
            m   = nm;
        }
        // combine (m, l, acc) across the 32 lanes of the wave
        for (int off = 16; off > 0; off >>= 1) {
            const float om = __shfl_xor(m,   off, 32);
            const float ol = __shfl_xor(l,   off, 32);
            const float oa = __shfl_xor(acc, off, 32);
            const float nm = fmaxf(m, om);
            const float s0 = __expf(m  - nm);
            const float s1 = __expf(om - nm);
            l   = l   * s0 + ol * s1;
            acc = acc * s0 + oa * s1;
            m   = nm;
        }
        if (lane == 0) s_ctx[b] = acc / l;
    }
    __syncthreads();

    // ---------------- Phase 2: 32x256 MLP as f32 WMMA GEMM ----------------
    // D[32x16] = H[32x256] x W2rep[256x16];  column N=0 = y.  Waves 0,1 = M-tiles.
    if (wave < 2) {
        const int   tile = wave;
        const int   hi   = lane >> 4;                      // lane half selects K pair
        const float ctxv = s_ctx[tile * 16 + (lane & 15)]; // M = lane % 16

        v8f c = {};
        for (int kk = 0; kk < MLP_HID; kk += 4) {
            const int   k0 = kk + 2 * hi;                  // A/B layout: VGPR r, half h -> K = r + 2h
            const int   k1 = k0 + 1;
            const float h0 = fmaxf(W1[k0] * ctxv + b1[k0], 0.f);
            const float h1 = fmaxf(W1[k1] * ctxv + b1[k1], 0.f);
            v2f A = {h0, h1};
            v2f B = {W2[k0], W2[k1]};                      // replicated across all N columns
            c = __builtin_amdgcn_wmma_f32_16x16x4_f32(
                    /*neg_a=*/false, A, /*neg_b=*/false, B,
                    /*c_mod=*/(short)0, c, /*reuse_a=*/false, /*reuse_b=*/false);
        }

        // D layout: lane (N=0) in {0,16}; VGPR r holds M = r + 8*hi
        if ((lane & 15) == 0) {
            const float bb    = b2[0];
            const int   mbase = tile * 16 + hi * 8;
            #pragma unroll
            for (int r = 0; r < 8; ++r) out[mbase + r] = c[r] + bb;
        }
    }
}

extern "C" void kernel_launch(void* const* d_in, const int* in_sizes, int n_in,
                              void* d_out, int out_size, void* d_ws, size_t ws_size,
                              hipStream_t stream) {
    const float* src  = (const float*)d_in[0];
    const float* Wq   = (const float*)d_in[1];
    const float* bq   = (const float*)d_in[2];
    const float* Wk   = (const float*)d_in[3];
    const float* bk   = (const float*)d_in[4];
    const float* Wv   = (const float*)d_in[5];
    const float* bv   = (const float*)d_in[6];
    const float* bias = (const float*)d_in[7];
    const float* W1   = (const float*)d_in[8];
    const float* b1   = (const float*)d_in[9];
    const float* W2   = (const float*)d_in[10];
    const float* b2   = (const float*)d_in[11];

    fused_attn_mlp_gfx1250<<<1, 1024, 0, stream>>>(
        src, Wq, bq, Wk, bk, Wv, bv, bias, W1, b1, W2, b2, (float*)d_out);
}